// C_LIF_22093311770854
// MI455X (gfx1250) — compile-verified
//
#include <hip/hip_runtime.h>
#include <stdint.h>

// LIF spiking-neuron scan: x [16, 8192, 200] f32 -> spikes [16, 8192, 200] f32.
// Memory-bound (~210 MB @ 23.3 TB/s => ~9us floor). Staged through LDS with the
// CDNA5 Tensor Data Mover (coalesced DMA tiles), computed in-place, stored back
// via TDM. Double-buffered, pipelined on TENSORcnt.

typedef unsigned int u32x4 __attribute__((ext_vector_type(4)));
typedef int          i32x8 __attribute__((ext_vector_type(8)));
typedef int          i32x4 __attribute__((ext_vector_type(4)));

#define T_TOTAL 200
#define TC      50                 // time steps per chunk
#define NROWS   256                // neurons per workgroup (one per thread)
#define CHUNK_ELEMS (NROWS * TC)   // 12800 f32 = 50 KB per buffer

// Build a 2-D TDM descriptor: tile = [NROWS rows x TC dwords], row stride in
// global memory = T_TOTAL dwords. global_addr points at the tile start; tensor
// dims are set equal to the tile so no OOB path is ever taken.
__device__ __forceinline__ void tdm_desc(u32x4& g0, i32x8& g1,
                                         unsigned lds_off, const void* gptr) {
  unsigned long long ga = (unsigned long long)(uintptr_t)gptr;
  g0[0] = 1u;                                                 // count=1 (valid user D#)
  g0[1] = lds_off;                                            // lds_addr (bytes)
  g0[2] = (unsigned)ga;                                       // global_addr[31:0]
  g0[3] = (unsigned)((ga >> 32) & 0x01FFFFFFu) | (2u << 30);  // addr[56:32] | type=2
  g1[0] = (2 << 16);                                          // data_size=4B; mask=0; no pad
  g1[1] = (TC & 0xFFFF) << 16;                                // tensor_dim0[15:0]
  g1[2] = ((NROWS & 0xFFFF) << 16);                           // tensor_dim0 hi=0 | tensor_dim1 lo
  g1[3] = ((TC & 0xFFFF) << 16);                              // tensor_dim1 hi=0 | tile_dim0=TC
  g1[4] = (NROWS & 0xFFFF);                                   // tile_dim1=NROWS | tile_dim2=0
  g1[5] = T_TOTAL;                                            // tensor_dim0_stride lo (elems)
  g1[6] = 0;                                                  // stride hi | dim1_stride lo
  g1[7] = 0;                                                  // dim1_stride hi
}

__device__ __forceinline__ void tdm_load(unsigned lds_off, const void* gptr) {
  u32x4 g0; i32x8 g1; i32x4 z4 = (i32x4)0; i32x8 z8 = (i32x8)0;
  tdm_desc(g0, g1, lds_off, gptr);
  __builtin_amdgcn_tensor_load_to_lds(g0, g1, z4, z4, z8, 0);
}

__device__ __forceinline__ void tdm_store(unsigned lds_off, const void* gptr) {
  u32x4 g0; i32x8 g1; i32x4 z4 = (i32x4)0; i32x8 z8 = (i32x8)0;
  tdm_desc(g0, g1, lds_off, gptr);
  __builtin_amdgcn_tensor_store_from_lds(g0, g1, z4, z4, z8, 0);
}

// One 50-step chunk of the recurrence on this thread's row, in place:
// reads x[t], overwrites with spike o[t]. Row pitch 50 dwords -> 50*l mod 64
// is distinct over l=0..31 -> bank-conflict-free in wave32.
__device__ __forceinline__ void lif_chunk(float* __restrict__ row,
                                          float& M, float& S, float& E, float& o,
                                          float dm, float dsc) {
#pragma unroll
  for (int j = 0; j < TC; ++j) {
    float xi = row[j];
    M = dm  * (M + xi);
    S = dsc * (S + xi);
    E = dm * E + o * 0.5f;          // o_prev * VTH
    float u = M - S - E;
    o = (u > 0.5f) ? 1.0f : 0.0f;   // spike(u - VTH): exact predicate
    row[j] = o;
  }
}

__global__ __launch_bounds__(256)
void lif_scan_kernel(const float* __restrict__ x,
                     const float* __restrict__ dm_p,
                     const float* __restrict__ ds_p,
                     float* __restrict__ out) {
  __shared__ __align__(16) float tile[2][CHUNK_ELEMS];   // 100 KB LDS

  const int  tid = threadIdx.x;
  const bool w0  = (tid >> 5) == 0;                      // wave 0 (uniform per wave)
  const long long nbase = (long long)blockIdx.x * NROWS;
  const float* gx = x   + nbase * T_TOTAL;
  float*       go = out + nbase * T_TOTAL;
  const unsigned lds0 = (unsigned)(uintptr_t)&tile[0][0];
  const unsigned lds1 = (unsigned)(uintptr_t)&tile[1][0];

  if (w0) {                                              // prologue: prefetch 2 chunks
    tdm_load(lds0, gx + 0 * TC);
    tdm_load(lds1, gx + 1 * TC);
  }

  const float dm  = dm_p[0];
  const float dsc = ds_p[0];
  float M = 0.f, S = 0.f, E = 0.f, o = 0.f;
  float* row0 = &tile[0][tid * TC];
  float* row1 = &tile[1][tid * TC];

  // Same-wave TDM ops complete in order -> exact tensorcnt waits.
  // Queue: L0 L1 | S0 L2 | S1 L3 | S2 | S3

  // chunk 0 (buf 0): need L0 done; L1 may remain -> wait <= 1
  if (w0) __builtin_amdgcn_s_wait_tensorcnt(1);
  __syncthreads();
  lif_chunk(row0, M, S, E, o, dm, dsc);
  __syncthreads();
  if (w0) { tdm_store(lds0, go + 0 * TC); tdm_load(lds0, gx + 2 * TC); }

  // chunk 1 (buf 1): need L1 done; {S0, L2} may remain -> wait <= 2
  if (w0) __builtin_amdgcn_s_wait_tensorcnt(2);
  __syncthreads();
  lif_chunk(row1, M, S, E, o, dm, dsc);
  __syncthreads();
  if (w0) { tdm_store(lds1, go + 1 * TC); tdm_load(lds1, gx + 3 * TC); }

  // chunk 2 (buf 0): need L2 done; {S1, L3} may remain -> wait <= 2
  if (w0) __builtin_amdgcn_s_wait_tensorcnt(2);
  __syncthreads();
  lif_chunk(row0, M, S, E, o, dm, dsc);
  __syncthreads();
  if (w0) { tdm_store(lds0, go + 2 * TC); }

  // chunk 3 (buf 1): need L3 done; {S2} may remain -> wait <= 1
  if (w0) __builtin_amdgcn_s_wait_tensorcnt(1);
  __syncthreads();
  lif_chunk(row1, M, S, E, o, dm, dsc);
  __syncthreads();
  if (w0) {
    tdm_store(lds1, go + 3 * TC);
    __builtin_amdgcn_s_wait_tensorcnt(0);                // drain before LDS dealloc
  }
}

extern "C" void kernel_launch(void* const* d_in, const int* in_sizes, int n_in,
                              void* d_out, int out_size, void* d_ws, size_t ws_size,
                              hipStream_t stream) {
  const float* x  = (const float*)d_in[0];
  const float* dm = (const float*)d_in[1];   // decay_m scalar (1 elem)
  const float* ds = (const float*)d_in[2];   // decay_s scalar (1 elem)
  float* out = (float*)d_out;

  const long long neurons = (long long)in_sizes[0] / T_TOTAL;  // 16*8192 = 131072
  const int blocks = (int)(neurons / NROWS);                   // 512
  lif_scan_kernel<<<blocks, 256, 0, stream>>>(x, dm, ds, out);
}